// CRF_10642928960085
// MI455X (gfx1250) — compile-verified
//
#include <hip/hip_runtime.h>
#include <stdint.h>

#define NTAGS 16
#define SEQL  512
#define START_TAG 1
#define END_TAG 2
#define CHUNK 64                 // timesteps per TDM chunk
#define NCHUNK (SEQL / CHUNK)
#define WAVES 8                  // waves per block (block = 256 threads)

typedef float v2f  __attribute__((ext_vector_type(2)));
typedef float v8f  __attribute__((ext_vector_type(8)));
typedef unsigned int u32x4 __attribute__((ext_vector_type(4)));
typedef unsigned int u32x8 __attribute__((ext_vector_type(8)));

// SWAPX16: swap lanes 0-15 <-> 16-31 (group-of-32 swizzle, xor=0x10, and=0x1f)
__device__ __forceinline__ float swz16(float x) {
  return __int_as_float(__builtin_amdgcn_ds_swizzle(__float_as_int(x), 0x401F));
}

// ---- Tensor Data Mover: contiguous NEL f32 global -> LDS ----
template <unsigned NEL>
__device__ __forceinline__ void tdm_copy(const float* gsrc, unsigned lds_byte_off) {
  unsigned glo  = __builtin_amdgcn_readfirstlane((unsigned)(uintptr_t)gsrc);
  unsigned ghi  = __builtin_amdgcn_readfirstlane((unsigned)(((unsigned long long)(uintptr_t)gsrc) >> 32));
  unsigned loff = __builtin_amdgcn_readfirstlane(lds_byte_off);
  u32x4 g0;
  g0.x = 1u;                                       // count=1, user-mode descriptor
  g0.y = loff;                                     // lds_addr (bytes)
  g0.z = glo;                                      // global_addr[31:0]
  g0.w = (ghi & 0x01FFFFFFu) | (2u << 30);         // global_addr[56:32] | type=2
  u32x8 g1;
  g1.s0 = 0x00020000u;                             // wg_mask=0, data_size=2 (4B)
  g1.s1 = (NEL & 0xFFFFu) << 16;                   // tensor_dim0[15:0]
  g1.s2 = (NEL >> 16) | (1u << 16);                // tensor_dim0[31:16] | tensor_dim1=1
  g1.s3 = (NEL & 0xFFFFu) << 16;                   // tile_dim0=NEL
  g1.s4 = 1u;                                      // tile_dim1=1, tile_dim2=0
  g1.s5 = NEL;                                     // tensor_dim0_stride[31:0]
  g1.s6 = 0u;
  g1.s7 = 0u;
  u32x4 g2; g2.x = 1u; g2.y = 0u; g2.z = 0u; g2.w = 0u;
  u32x4 g3; g3.x = 0u; g3.y = 0u; g3.z = 0u; g3.w = 0u;
  asm volatile("tensor_load_to_lds %0, %1, %2, %3"
               :: "s"(g0), "s"(g1), "s"(g2), "s"(g3)
               : "memory");
}

__device__ __forceinline__ void wait_tensorcnt0() {
  asm volatile("s_wait_tensorcnt 0x0" ::: "memory");
}
__device__ __forceinline__ void wait_tensorcnt2() {
  asm volatile("s_wait_tensorcnt 0x2" ::: "memory");
}

// ---- Forward DP: one wave per batch; Q = P^T chained via WMMA f32 ----
__global__ __launch_bounds__(256) void crf_dp_wmma(
    const float* __restrict__ em, const float* __restrict__ mask,
    const float* __restrict__ trans, const float* __restrict__ score,
    float* __restrict__ diff)
{
  __shared__ float emsm[WAVES][2][CHUNK * NTAGS];  // 64KB emission tiles
  __shared__ float msm[WAVES][2][CHUNK];           // 4KB mask tiles
  const int lane = threadIdx.x & 31;
  const int wid  = threadIdx.x >> 5;
  const int b    = blockIdx.x * WAVES + wid;
  const int h    = lane >> 4;       // wave half
  const int j    = lane & 15;       // matrix column (A row index / D col index)

  // A = exp(T)^T in WMMA A-layout (16x4 chunks): reg 2k+r holds A[j][i], i=4k+r+2h
  float a[8], et[8], d[8];
#pragma unroll
  for (int k = 0; k < 4; ++k)
#pragma unroll
    for (int r = 0; r < 2; ++r) {
      const int i = 4 * k + r + 2 * h;
      a[2 * k + r] = __expf(trans[i * NTAGS + j]);     // exp(T[i][j])
    }
#pragma unroll
  for (int v = 0; v < 8; ++v) {
    et[v] = __expf(trans[(v + 8 * h) * NTAGS + END_TAG]);
    d[v]  = ((v + 8 * h) == j) ? 1.0f : 0.0f;          // Q = Identity
  }

  int ktot = 0;  // accumulated power-of-two rescale exponent
  const float* emb = em   + (size_t)b * SEQL * NTAGS;
  const float* mb  = mask + (size_t)b * SEQL;
  const unsigned elds0 = (unsigned)(uintptr_t)(&emsm[wid][0][0]);
  const unsigned elds1 = (unsigned)(uintptr_t)(&emsm[wid][1][0]);
  const unsigned mlds0 = (unsigned)(uintptr_t)(&msm[wid][0][0]);
  const unsigned mlds1 = (unsigned)(uintptr_t)(&msm[wid][1][0]);

  tdm_copy<CHUNK * NTAGS>(emb, elds0);
  tdm_copy<CHUNK>(mb, mlds0);
  for (int c = 0; c < NCHUNK; ++c) {
    if (c + 1 < NCHUNK) {
      const int nb = (c + 1) & 1;
      tdm_copy<CHUNK * NTAGS>(emb + (size_t)(c + 1) * CHUNK * NTAGS, nb ? elds1 : elds0);
      tdm_copy<CHUNK>(mb + (size_t)(c + 1) * CHUNK, nb ? mlds1 : mlds0);
      wait_tensorcnt2();            // chunk c ready, chunk c+1 in flight
    } else {
      wait_tensorcnt0();
    }
    const float* buf  = emsm[wid][c & 1];
    const float* mbuf = msm[wid][c & 1];
#pragma unroll 8
    for (int tl = 0; tl < CHUNK; ++tl) {
      const int   t   = c * CHUNK + tl;
      const float m   = mbuf[tl];                  // broadcast ds_load
      const float emv = buf[tl * NTAGS + j];
      const float s   = __expf(emv);               // row scale exp(em_t[j])
      // D-layout Q -> B-layout (8 SWAPX16 swizzles + 8 selects)
      float bb[8];
      {
        const float x0 = swz16(d[0]), x1 = swz16(d[1]);
        const float x2 = swz16(d[2]), x3 = swz16(d[3]);
        const float x4 = swz16(d[4]), x5 = swz16(d[5]);
        const float x6 = swz16(d[6]), x7 = swz16(d[7]);
        bb[0] = h ? x2   : d[0];
        bb[1] = h ? x3   : d[1];
        bb[2] = h ? x6   : d[4];
        bb[3] = h ? x7   : d[5];
        bb[4] = h ? d[2] : x0;
        bb[5] = h ? d[3] : x1;
        bb[6] = h ? d[6] : x4;
        bb[7] = h ? d[7] : x5;
      }
      // Two independent 2-deep WMMA chains (halve the serial RAW depth)
      v8f acc1 = {}, acc2 = {};
      {
        v2f A0; A0.x = a[0] * s; A0.y = a[1] * s;
        v2f B0; B0.x = bb[0];    B0.y = bb[1];
        acc1 = __builtin_amdgcn_wmma_f32_16x16x4_f32(false, A0, false, B0, (short)0, acc1, false, false);
        v2f A1; A1.x = a[2] * s; A1.y = a[3] * s;
        v2f B1; B1.x = bb[2];    B1.y = bb[3];
        acc1 = __builtin_amdgcn_wmma_f32_16x16x4_f32(false, A1, false, B1, (short)0, acc1, false, false);
        v2f A2; A2.x = a[4] * s; A2.y = a[5] * s;
        v2f B2; B2.x = bb[4];    B2.y = bb[5];
        acc2 = __builtin_amdgcn_wmma_f32_16x16x4_f32(false, A2, false, B2, (short)0, acc2, false, false);
        v2f A3; A3.x = a[6] * s; A3.y = a[7] * s;
        v2f B3; B3.x = bb[6];    B3.y = bb[7];
        acc2 = __builtin_amdgcn_wmma_f32_16x16x4_f32(false, A3, false, B3, (short)0, acc2, false, false);
      }
      // Branchless mask blend: EXEC stays all-ones for every WMMA
      const bool on = (m != 0.0f);
#pragma unroll
      for (int v = 0; v < 8; ++v) {
        const float nv = acc1[v] + acc2[v];
        d[v] = on ? nv : d[v];
      }
      if ((t & 7) == 7) {           // exact power-of-two renormalization
        float gm = d[0];
#pragma unroll
        for (int v = 1; v < 8; ++v) gm = fmaxf(gm, d[v]);
        gm = fmaxf(gm, __shfl_xor(gm, 1, 32));
        gm = fmaxf(gm, __shfl_xor(gm, 2, 32));
        gm = fmaxf(gm, __shfl_xor(gm, 4, 32));
        gm = fmaxf(gm, __shfl_xor(gm, 8, 32));
        gm = fmaxf(gm, __shfl_xor(gm, 16, 32));
        int e = (int)((__float_as_uint(gm) >> 23) & 0xFFu) - 127;
        e = (e < -100) ? 0 : ((e > 126) ? 126 : e);
        const float sc = __uint_as_float((unsigned)(127 - e) << 23); // 2^-e
#pragma unroll
        for (int v = 0; v < 8; ++v) d[v] *= sc;
        ktot += e;
      }
    }
  }

  // dp = log( sum_j (sum_i Q[j][i]) * exp(T[j][END]) ) + ktot*ln2
  float rs[8];
#pragma unroll
  for (int v = 0; v < 8; ++v) {
    float r = d[v];
    r += __shfl_xor(r, 1, 32);
    r += __shfl_xor(r, 2, 32);
    r += __shfl_xor(r, 4, 32);
    r += __shfl_xor(r, 8, 32);
    rs[v] = r;                      // row sums for rows v+8h
  }
  float s2 = 0.0f;
#pragma unroll
  for (int v = 0; v < 8; ++v) s2 = fmaf(rs[v], et[v], s2);
  s2 += __shfl_xor(s2, 16, 32);
  const float dp = __logf(s2) + (float)ktot * 0.6931471805599453f;
  if (lane == 0) diff[b] = score[b] - dp;
}

// ---- Gold path score: one wave per batch, lanes stride over timesteps ----
__global__ __launch_bounds__(256) void crf_score(
    const float* __restrict__ em, const int* __restrict__ tags,
    const float* __restrict__ mask, const float* __restrict__ trans,
    float* __restrict__ score_out)
{
  __shared__ float tsm[NTAGS * NTAGS];
  if (threadIdx.x < NTAGS * NTAGS) tsm[threadIdx.x] = trans[threadIdx.x];
  __syncthreads();
  const int lane = threadIdx.x & 31;
  const int wid  = threadIdx.x >> 5;
  const int b    = blockIdx.x * WAVES + wid;
  const int*   tg  = tags + (size_t)b * SEQL;
  const float* emb = em   + (size_t)b * SEQL * NTAGS;
  const float* mb  = mask + (size_t)b * SEQL;
  float acc = 0.0f, msum = 0.0f;
  for (int t = lane; t < SEQL; t += 32) {
    const int   ct = tg[t];
    const float m  = mb[t];
    msum += m;
    if (t == 0) {
      acc += tsm[START_TAG * NTAGS + ct] + emb[ct];
    } else {
      const int pt = tg[t - 1];
      acc += (tsm[pt * NTAGS + ct] + emb[(size_t)t * NTAGS + ct]) * m;
    }
  }
  acc  += __shfl_xor(acc, 1, 32);  msum += __shfl_xor(msum, 1, 32);
  acc  += __shfl_xor(acc, 2, 32);  msum += __shfl_xor(msum, 2, 32);
  acc  += __shfl_xor(acc, 4, 32);  msum += __shfl_xor(msum, 4, 32);
  acc  += __shfl_xor(acc, 8, 32);  msum += __shfl_xor(msum, 8, 32);
  acc  += __shfl_xor(acc, 16, 32); msum += __shfl_xor(msum, 16, 32);
  int last = (int)msum - 1;
  last = last < 0 ? 0 : last;
  const int lt = tg[last];
  if (lane == 0) score_out[b] = acc + tsm[lt * NTAGS + END_TAG];
}

// ---- Deterministic final reduction: out = -sum(diff) ----
__global__ __launch_bounds__(256) void crf_reduce(
    const float* __restrict__ diff, float* __restrict__ out, int n)
{
  __shared__ float sm[256];
  float s = 0.0f;
  for (int i = threadIdx.x; i < n; i += 256) s += diff[i];
  sm[threadIdx.x] = s;
  __syncthreads();
  for (int off = 128; off > 0; off >>= 1) {
    if ((int)threadIdx.x < off) sm[threadIdx.x] += sm[threadIdx.x + off];
    __syncthreads();
  }
  if (threadIdx.x == 0) out[0] = -sm[0];
}

extern "C" void kernel_launch(void* const* d_in, const int* in_sizes, int n_in,
                              void* d_out, int out_size, void* d_ws, size_t ws_size,
                              hipStream_t stream) {
  const float* em    = (const float*)d_in[0];
  const int*   tags  = (const int*)d_in[1];
  const float* mask  = (const float*)d_in[2];
  const float* trans = (const float*)d_in[3];
  float*       out   = (float*)d_out;
  const int bs = in_sizes[2] / SEQL;          // mask has bs*SEQL elements

  float* ws_score = (float*)d_ws;             // [bs]
  float* ws_diff  = ws_score + bs;            // [bs]

  const int blocks = bs / WAVES;
  crf_score<<<blocks, 256, 0, stream>>>(em, tags, mask, trans, ws_score);
  crf_dp_wmma<<<blocks, 256, 0, stream>>>(em, mask, trans, ws_score, ws_diff);
  crf_reduce<<<1, 256, 0, stream>>>(ws_diff, out, bs);
}